// ConsistenceLoss_33234456937041
// MI455X (gfx1250) — compile-verified
//
#include <hip/hip_runtime.h>
#include <stdint.h>

#define B_    32
#define T_    2048
#define D_    512
#define MAXR  1025
#define NS_   (B_ * MAXR)        // 32800 runs max
#define TILE  32
#define NTILE (T_ / TILE)        // 64 tiles per row
#define NSLOT (B_ * NTILE)       // 2048 boundary-run slots
#define DQ    128                // features per workgroup in feat kernel
#define CPTf  0.5f
#define CTf   0.7f
#define FWf   1.0f
#define AWf   1.0f

typedef float v2f __attribute__((ext_vector_type(2)));
typedef float v8f __attribute__((ext_vector_type(8)));
typedef int   v4i __attribute__((ext_vector_type(4)));
typedef __attribute__((address_space(3))) v4i* lds_v4i_ptr;

#if __has_builtin(__builtin_amdgcn_global_load_async_to_lds_b128)
#define HAVE_ASYNC_LDS 1
#else
#define HAVE_ASYNC_LDS 0
#endif

// ---------------------------------------------------------------------------
// Kernel 1: per batch row — run-length encode attn, per-run stats, attn loss.
// One 256-thread block per row; 8 consecutive positions per thread.
// ---------------------------------------------------------------------------
__global__ __launch_bounds__(256) void attn_kernel(
    const float* __restrict__ attn, uint32_t* __restrict__ segw,
    float* __restrict__ cntArr, float* __restrict__ repArr,
    int* __restrict__ slotArr, int* __restrict__ slotRun,
    float* __restrict__ vidloss)
{
  __shared__ float aS[T_];                  // 8 KB
  __shared__ int   scan[256];
  __shared__ float cnt[MAXR], asum[MAXR], repc[MAXR], msum[MAXR];
  __shared__ int   spos[MAXR], epos[MAXR];
  __shared__ int   nruns_s;
  __shared__ float red[256];

  const int b = blockIdx.x;
  const int t = threadIdx.x;

  for (int p = t; p < T_; p += 256) aS[p] = attn[b * T_ + p];
  for (int r = t; r < MAXR; r += 256) { cnt[r] = 0.f; asum[r] = 0.f; repc[r] = 0.f; msum[r] = 0.f; }
  __syncthreads();

  const int p0 = t * 8;
  int starts = 0;
#pragma unroll
  for (int j = 0; j < 8; ++j) {
    int p = p0 + j;
    bool pr = aS[p] > CPTf;
    bool pv = (p > 0) && (aS[p - 1] > CPTf);
    starts += (pr && !pv) ? 1 : 0;
  }
  scan[t] = starts;
  __syncthreads();
  for (int offs = 1; offs < 256; offs <<= 1) {   // Hillis-Steele inclusive scan
    int v = scan[t];
    int add = (t >= offs) ? scan[t - offs] : 0;
    __syncthreads();
    scan[t] = v + add;
    __syncthreads();
  }
  const int excl = scan[t] - starts;
  if (t == 255) nruns_s = scan[255];
  __syncthreads();
  const int nruns = nruns_s;

  // Pass A: seg words + per-run count/sum/rep/start/end
  int rid = excl - 1;
#pragma unroll
  for (int j = 0; j < 8; ++j) {
    int p = p0 + j;
    float av = aS[p];
    bool pr = av > CPTf;
    bool pv = (p > 0) && (aS[p - 1] > CPTf);
    bool st = pr && !pv;
    bool rp = av > CTf;
    if (st) rid++;
    uint32_t wd = ((uint32_t)(rid & 0xFFFF)) | (pr ? (1u << 16) : 0u) | (rp ? (1u << 17) : 0u);
    segw[b * T_ + p] = wd;
    if (pr) {
      atomicAdd(&cnt[rid], 1.f);
      atomicAdd(&asum[rid], av);
      if (rp) atomicAdd(&repc[rid], 1.f);
      if (st) spos[rid] = p;
      if (p == T_ - 1 || !(aS[p + 1] > CPTf)) epos[rid] = p;
    }
  }
  __syncthreads();

  // Pass B: per-run attn variance (2-pass MSE like the reference)
  rid = excl - 1;
#pragma unroll
  for (int j = 0; j < 8; ++j) {
    int p = p0 + j;
    float av = aS[p];
    bool pr = av > CPTf;
    bool pv = (p > 0) && (aS[p - 1] > CPTf);
    if (pr && !pv) rid++;
    if (pr) {
      float m = asum[rid] / cnt[rid];
      float d = av - m;
      atomicAdd(&msum[rid], d * d);
    }
  }
  __syncthreads();

  // Per-run global outputs + vid_loss
  float vsum = 0.f;
  for (int r = t; r < nruns; r += 256) {
    int glob = b * MAXR + r;
    cntArr[glob] = cnt[r];
    repArr[glob] = repc[r];
    int stile = spos[r] / TILE, etile = epos[r] / TILE;
    if (etile != stile) {                 // crosses >=1 tile boundary
      int slot = b * NTILE + stile + 1;   // unique: one crossing run per start-tile
      slotArr[glob] = slot;
      slotRun[slot] = glob;
    }
    vsum += msum[r] / cnt[r];
  }
  red[t] = vsum;
  __syncthreads();
  for (int s = 128; s; s >>= 1) { if (t < s) red[t] += red[t + s]; __syncthreads(); }
  if (t == 0) vidloss[b] = red[0] / (float)(nruns > 0 ? nruns : 1);
}

// ---------------------------------------------------------------------------
// Kernel 2 (heavy, HBM-bound): segmented feature sums per run via exact-fp32
// WMMA 16x16x4_f32, with the feat tile staged into LDS by async
// GLOBAL_LOAD_ASYNC_TO_LDS_B128 copies (ASYNCcnt path) when available.
// Grid: (tile=64, dquarter=4, b=32); 256 threads = 8 waves, 16 features/wave.
// Per 16-position chunk: A=indicator(16 runs x 4 pos), B=feat(4 pos x 16
// feats) from LDS, C accumulates 16x16 fp32. Interior runs -> 3-scalar dot
// atomics; tile-crossing runs -> vector atomics into slot scratch.
// ---------------------------------------------------------------------------
__global__ __launch_bounds__(256) void feat_kernel(
    const float* __restrict__ feat, const uint32_t* __restrict__ segw,
    const int* __restrict__ slotArr,
    float* __restrict__ sww, float* __restrict__ swr, float* __restrict__ srr,
    float* __restrict__ vecW, float* __restrict__ vecR)
{
  __shared__ __align__(16) float tileF[TILE][DQ];   // 16 KB staged feat tile
  __shared__ float Sw[17][DQ];                       // 17 run buckets x 128 feats
  __shared__ float Sr[17][DQ];
  __shared__ uint32_t seg_s[TILE];

  const int tid  = threadIdx.x;
  const int lane = tid & 31;
  const int wave = tid >> 5;                // 0..7
  const int tileIdx = blockIdx.x;           // 0..63
  const int dq      = blockIdx.y;           // 0..3
  const int b       = blockIdx.z;           // 0..31
  const int t0 = tileIdx * TILE;

  // --- stage 32 rows x 32 fp32 (first 128B of each row) into LDS: one 16B chunk per thread ---
  {
    const int row = tid >> 3;               // 0..31
    const int ch  = tid & 7;                // 0..7 (16B chunks of the first 128B of a row)
    const size_t goff = ((size_t)b * T_ + (size_t)(t0 + row)) * D_
                      + (size_t)dq * DQ + (size_t)ch * 4;
#if HAVE_ASYNC_LDS
    __builtin_amdgcn_global_load_async_to_lds_b128(
        (v4i*)(const_cast<float*>(feat) + goff),
        (lds_v4i_ptr)&tileF[row][ch * 4],
        0, 0);
#else
    float4 v = *(const float4*)(feat + goff);
    *(float4*)&tileF[row][ch * 4] = v;
#endif
  }

  for (int i = tid; i < 17 * DQ; i += 256) {
    (&Sw[0][0])[i] = 0.f;
    (&Sr[0][0])[i] = 0.f;
  }
  if (tid < TILE) seg_s[tid] = segw[b * T_ + t0 + tid];

#if HAVE_ASYNC_LDS
#if __has_builtin(__builtin_amdgcn_s_wait_asynccnt)
  __builtin_amdgcn_s_wait_asynccnt(0);
#else
  asm volatile("s_wait_asynccnt 0" ::: "memory");
#endif
#endif
  __syncthreads();

  const int m    = lane & 15;               // A-matrix row (run-local id)
  const int kk   = (lane >> 4) * 2;         // K pair base for this lane half
  const int col  = wave * 16 + m;           // feature column within 128-slice

  int R0 = (int)(short)(seg_s[0] & 0xFFFFu);       // tile-base run id (may be -1)

  for (int c = 0; c < 2; ++c) {             // two 16-position chunks
    const int pbase = c * 16;
    const int baseRun = (int)(short)(seg_s[pbase] & 0xFFFFu);
    v8f cw = {0.f, 0.f, 0.f, 0.f, 0.f, 0.f, 0.f, 0.f};
    v8f cr = {0.f, 0.f, 0.f, 0.f, 0.f, 0.f, 0.f, 0.f};

    for (int k4 = 0; k4 < 4; ++k4) {        // K-loop: 4 positions per WMMA
      const int pc = pbase + k4 * 4;
      v2f aw, ar, bv;
#pragma unroll
      for (int j = 0; j < 2; ++j) {
        const int pl = pc + kk + j;         // position within tile for this lane/VGPR
        const uint32_t wd = seg_s[pl];
        const int rid = (int)(short)(wd & 0xFFFFu);
        const bool pr = (wd >> 16) & 1u;
        const bool rp = (wd >> 17) & 1u;
        const bool hit = (rid - baseRun) == m;
        const float onw = (pr && hit) ? 1.f : 0.f;
        const float onr = (rp && hit) ? 1.f : 0.f;
        const float fv = tileF[pl][col];    // B operand from staged LDS tile
        if (j == 0) { aw.x = onw; ar.x = onr; bv.x = fv; }
        else        { aw.y = onw; ar.y = onr; bv.y = fv; }
      }
      cw = __builtin_amdgcn_wmma_f32_16x16x4_f32(false, aw, false, bv, (short)0, cw, false, false);
      cr = __builtin_amdgcn_wmma_f32_16x16x4_f32(false, ar, false, bv, (short)0, cr, false, false);
    }

    // Flush C into LDS buckets, rebased to the tile-base run.
    const int off = baseRun - R0;
#pragma unroll
    for (int i = 0; i < 8; ++i) {
      const int mrow = i + 8 * (lane >> 4);       // C row M for this lane/VGPR
      const int bkt = off + mrow;
      if (bkt >= 0 && bkt < 17) {
        Sw[bkt][col] += cw[i];
        Sr[bkt][col] += cr[i];
      }
    }
  }
  __syncthreads();

  // Finalize buckets: wave w handles buckets w, w+8, w+16.
  for (int j = wave; j < 17; j += 8) {
    const int g = R0 + j;
    if (g < 0 || g >= MAXR) continue;
    const int glob = b * MAXR + g;
    const int slot = slotArr[glob];
    if (slot < 0) {
      // interior run: partial dot products over this WG's 128 features
      float dww = 0.f, dwr = 0.f, drr = 0.f;
      for (int f = lane; f < DQ; f += 32) {
        float x = Sw[j][f], y = Sr[j][f];
        dww += x * x; dwr += x * y; drr += y * y;
      }
      for (int s = 16; s > 0; s >>= 1) {
        dww += __shfl_xor(dww, s, 32);
        dwr += __shfl_xor(dwr, s, 32);
        drr += __shfl_xor(drr, s, 32);
      }
      if (lane == 0 && (dww != 0.f || drr != 0.f)) {
        atomicAdd(&sww[glob], dww);
        atomicAdd(&swr[glob], dwr);
        atomicAdd(&srr[glob], drr);
      }
    } else {
      // crossing run: accumulate partial vectors into slot scratch
      for (int f = lane; f < DQ; f += 32) {
        float x = Sw[j][f], y = Sr[j][f];
        size_t base = (size_t)slot * D_ + (size_t)dq * DQ + f;
        if (x != 0.f) atomicAdd(&vecW[base], x);
        if (y != 0.f) atomicAdd(&vecR[base], y);
      }
    }
  }
}

// ---------------------------------------------------------------------------
// Kernel 3a: finalize tile-crossing runs — dot the accumulated 512-vectors.
// ---------------------------------------------------------------------------
__global__ __launch_bounds__(256) void slot_kernel(
    const float* __restrict__ vecW, const float* __restrict__ vecR,
    const int* __restrict__ slotRun,
    float* __restrict__ sww, float* __restrict__ swr, float* __restrict__ srr)
{
  __shared__ float r0[8], r1[8], r2[8];
  const int slot = blockIdx.x;
  const int g = slotRun[slot];
  if (g < 0) return;                      // uniform per block
  const int t = threadIdx.x;
  float dww = 0.f, dwr = 0.f, drr = 0.f;
  for (int f = t; f < D_; f += 256) {
    float x = vecW[(size_t)slot * D_ + f];
    float y = vecR[(size_t)slot * D_ + f];
    dww += x * x; dwr += x * y; drr += y * y;
  }
  for (int s = 16; s > 0; s >>= 1) {
    dww += __shfl_xor(dww, s, 32);
    dwr += __shfl_xor(dwr, s, 32);
    drr += __shfl_xor(drr, s, 32);
  }
  const int wv = t >> 5, ln = t & 31;
  if (ln == 0) { r0[wv] = dww; r1[wv] = dwr; r2[wv] = drr; }
  __syncthreads();
  if (t == 0) {
    float a = 0.f, bb = 0.f, c = 0.f;
    for (int i = 0; i < 8; ++i) { a += r0[i]; bb += r1[i]; c += r2[i]; }
    atomicAdd(&sww[g], a);
    atomicAdd(&swr[g], bb);
    atomicAdd(&srr[g], c);
  }
}

// ---------------------------------------------------------------------------
// Kernel 3b: final reduction over all runs -> scalar loss.
// ---------------------------------------------------------------------------
__global__ __launch_bounds__(1024) void final_kernel(
    const float* __restrict__ cntArr, const float* __restrict__ repArr,
    const float* __restrict__ sww, const float* __restrict__ swr,
    const float* __restrict__ srr, const float* __restrict__ vidloss,
    float* __restrict__ out)
{
  __shared__ float fred[1024];
  __shared__ int   ired[1024];
  const int t = threadIdx.x;
  float acc = 0.f; int nq = 0;
  for (int g = t; g < NS_; g += 1024) {
    float c = cntArr[g], r = repArr[g];
    if (c > 0.f && r > 0.f) {             // valid run containing a rep
      float mse = (sww[g] / (c * c) - 2.f * swr[g] / (c * r) + srr[g] / (r * r)) * (1.0f / (float)D_);
      acc += mse; nq += 1;
    }
  }
  fred[t] = acc; ired[t] = nq;
  __syncthreads();
  for (int s = 512; s; s >>= 1) {
    if (t < s) { fred[t] += fred[t + s]; ired[t] += ired[t + s]; }
    __syncthreads();
  }
  if (t == 0) {
    float attn = 0.f;
    for (int b = 0; b < B_; ++b) attn += vidloss[b];
    attn *= (1.0f / (float)B_);
    int nqT = ired[0] > 0 ? ired[0] : 1;
    out[0] = FWf * (fred[0] / (float)nqT) + AWf * attn;
  }
}

// ---------------------------------------------------------------------------
extern "C" void kernel_launch(void* const* d_in, const int* in_sizes, int n_in,
                              void* d_out, int out_size, void* d_ws, size_t ws_size,
                              hipStream_t stream)
{
  const float* attn = (const float*)d_in[0];   // [32,2048,1]
  const float* feat = (const float*)d_in[1];   // [32,2048,512]
  float* out = (float*)d_out;
  char* ws = (char*)d_ws;

  size_t off = 0;
  auto take = [&](size_t bytes) { size_t o = off; off = (off + bytes + 255) & ~(size_t)255; return o; };

  // Zero-initialized region (contiguous)
  const size_t o_cnt  = take((size_t)NS_ * 4);
  const size_t o_rep  = take((size_t)NS_ * 4);
  const size_t o_sww  = take((size_t)NS_ * 4);
  const size_t o_swr  = take((size_t)NS_ * 4);
  const size_t o_srr  = take((size_t)NS_ * 4);
  const size_t o_vid  = take((size_t)B_ * 4);
  const size_t o_vecW = take((size_t)NSLOT * D_ * 4);
  const size_t o_vecR = take((size_t)NSLOT * D_ * 4);
  const size_t zero_end = off;
  // 0xFF-initialized region (ints -> -1)
  const size_t o_slotA = take((size_t)NS_ * 4);
  const size_t o_slotR = take((size_t)NSLOT * 4);
  const size_t ff_end = off;
  // No-init region (fully written)
  const size_t o_seg = take((size_t)B_ * T_ * 4);
  (void)o_seg; (void)ws_size; (void)in_sizes; (void)n_in; (void)out_size;

  float*    cntArr  = (float*)(ws + o_cnt);
  float*    repArr  = (float*)(ws + o_rep);
  float*    swwA    = (float*)(ws + o_sww);
  float*    swrA    = (float*)(ws + o_swr);
  float*    srrA    = (float*)(ws + o_srr);
  float*    vidloss = (float*)(ws + o_vid);
  float*    vecW    = (float*)(ws + o_vecW);
  float*    vecR    = (float*)(ws + o_vecR);
  int*      slotArr = (int*)(ws + o_slotA);
  int*      slotRun = (int*)(ws + o_slotR);
  uint32_t* segw    = (uint32_t*)(ws + o_seg);

  (void)hipMemsetAsync(ws, 0, zero_end, stream);
  (void)hipMemsetAsync(ws + o_slotA, 0xFF, ff_end - o_slotA, stream);

  attn_kernel<<<B_, 256, 0, stream>>>(attn, segw, cntArr, repArr, slotArr, slotRun, vidloss);
  feat_kernel<<<dim3(NTILE, 4, B_), 256, 0, stream>>>(feat, segw, slotArr, swwA, swrA, srrA, vecW, vecR);
  slot_kernel<<<NSLOT, 256, 0, stream>>>(vecW, vecR, slotRun, swwA, swrA, srrA);
  final_kernel<<<1, 1024, 0, stream>>>(cntArr, repArr, swwA, swrA, srrA, vidloss, out);
}